// POEMVPolicy_29841432773436
// MI455X (gfx1250) — compile-verified
//
#include <hip/hip_runtime.h>
#include <cstdint>

#define EPSV 1e-10f

typedef float v4f __attribute__((ext_vector_type(4)));

// Low 32 bits of a flat shared-aperture address are the raw LDS byte offset
// (ISA 10.2).
__device__ __forceinline__ uint32_t lds_addr_of(const void* p) {
    return (uint32_t)(uintptr_t)p;
}

// Per-element closed-form math for N=2.
__device__ __forceinline__ void poemv_core(
    float pk, float xk, float dk, float fk,
    float omega, float rr, float lam,
    float m10, float m11, float m20, float m21,
    float a00, float a01, float a11,
    float b00, float b01, float b11,
    float ph00, float ph01, float ph10, float ph11,
    float c11, float c12, float c22,
    float* mean2, float* cov4)
{
    float qk  = 1.0f - pk;
    // excess_mu = p*mu1 + (1-p)*mu2 - r
    float ex0 = pk * m10 + qk * m20 - rr;
    float ex1 = pk * m11 + qk * m21 - rr;
    // Sigma_bar (symmetric 2x2) + EPS*I
    float s11 = pk * a00 + qk * b00 + EPSV;
    float s12 = pk * a01 + qk * b01;
    float s22 = pk * a11 + qk * b11 + EPSV;
    // Cholesky:  U = chol(Sigma)^T = [[u11,u12],[0,u22]]
    float u11  = sqrtf(s11);
    float iu11 = 1.0f / u11;
    float u12  = s12 * iu11;
    float u22  = sqrtf(s22 - u12 * u12);
    float iu22 = 1.0f / u22;
    // signal and triangular solve U*sti = signal
    float adj  = pk - dk * pk * qk;
    float sg0  = ex0 * (ph00 * adj + ph01);
    float sg1  = ex1 * (ph10 * adj + ph11);
    float sti1 = sg1 * iu22;
    float sti0 = (sg0 - u12 * sti1) * iu11;
    float g    = omega - xk;           // -(x - omega)
    mean2[0] = g * sti0;
    mean2[1] = g * sti1;
    // inv_U = [[iu11, bi],[0, iu22]],  bi = -u12/(u11*u22)
    float bi = -u12 * iu11 * iu22;
    float cf = 0.5f * lam * fk;
    // cov = cf * inv_U * C * inv_U^T  (symmetric by construction) + EPS*I
    float t0 = iu11 * c11 + bi * c12;
    float t1 = iu11 * c12 + bi * c22;
    cov4[0] = cf * (t0 * iu11 + t1 * bi) + EPSV;
    float off = cf * (iu22 * t1);
    cov4[1] = off;
    cov4[2] = off;
    cov4[3] = cf * (iu22 * iu22 * c22) + EPSV;
}

__global__ __launch_bounds__(256) void poemv_vec_kernel(
    const float* __restrict__ x,  const float* __restrict__ p,
    const float* __restrict__ dl, const float* __restrict__ f,
    const float* __restrict__ omega_p,
    const float* __restrict__ mu1, const float* __restrict__ mu2,
    const float* __restrict__ S1,  const float* __restrict__ S2,
    const float* __restrict__ phi, const float* __restrict__ ctu,
    const int* __restrict__ lam_p, const int* __restrict__ r_p,
    float* __restrict__ mean_out, float* __restrict__ cov_out,
    int nvec)
{
    __shared__ v4f sx[2][256];
    __shared__ v4f sp[2][256];
    __shared__ v4f sd[2][256];
    __shared__ v4f sf[2][256];

    const int tid    = threadIdx.x;
    const int stride = gridDim.x * blockDim.x;
    int v = blockIdx.x * blockDim.x + tid;

    // Batch-invariant parameters (uniform -> scalar loads)
    const float omega = omega_p[0];
    const float rr    = (float)r_p[0];
    const float lam   = (float)lam_p[0];
    const float m10 = mu1[0], m11v = mu1[1];
    const float m20 = mu2[0], m21  = mu2[1];
    const float a00 = S1[0], a01 = S1[1], a11 = S1[3];
    const float b00 = S2[0], b01 = S2[1], b11 = S2[3];
    const float ph00 = phi[0], ph01 = phi[1], ph10 = phi[2], ph11 = phi[3];
    // spd_core = L L^T with L = tril(ctu), diag -> exp(diag)
    const float e0  = __expf(ctu[0]);
    const float e1  = __expf(ctu[3]);
    const float t10 = ctu[2];
    const float c11 = e0 * e0;
    const float c12 = e0 * t10;
    const float c22 = t10 * t10 + e1 * e1;

    auto issue = [&](int vv, int buf) {
        uint64_t gx = (uint64_t)(uintptr_t)((const v4f*)x  + vv);
        uint64_t gp = (uint64_t)(uintptr_t)((const v4f*)p  + vv);
        uint64_t gd = (uint64_t)(uintptr_t)((const v4f*)dl + vv);
        uint64_t gf = (uint64_t)(uintptr_t)((const v4f*)f  + vv);
        uint32_t lx = lds_addr_of(&sx[buf][tid]);
        uint32_t lp = lds_addr_of(&sp[buf][tid]);
        uint32_t ld = lds_addr_of(&sd[buf][tid]);
        uint32_t lf = lds_addr_of(&sf[buf][tid]);
        // 512B contiguous per wave per instruction; tracked with ASYNCcnt.
        asm volatile("global_load_async_to_lds_b128 %0, %1, off" :: "v"(lx), "v"(gx) : "memory");
        asm volatile("global_load_async_to_lds_b128 %0, %1, off" :: "v"(lp), "v"(gp) : "memory");
        asm volatile("global_load_async_to_lds_b128 %0, %1, off" :: "v"(ld), "v"(gd) : "memory");
        asm volatile("global_load_async_to_lds_b128 %0, %1, off" :: "v"(lf), "v"(gf) : "memory");
    };

    bool valid = v < nvec;
    int buf = 0;
    if (valid) issue(v, 0);

    while (valid) {
        int vn = v + stride;
        bool nvalid = vn < nvec;
        if (nvalid) {
            issue(vn, buf ^ 1);
            // 4 newest loads may remain outstanding; oldest 4 (current stage)
            // are guaranteed complete (async loads complete in order).
            asm volatile("s_wait_asynccnt 4" ::: "memory");
        } else {
            asm volatile("s_wait_asynccnt 0" ::: "memory");
        }

        v4f xv = sx[buf][tid];
        v4f pv = sp[buf][tid];
        v4f dv = sd[buf][tid];
        v4f fv = sf[buf][tid];

        float mean8[8];
        float cov16[16];
#pragma unroll
        for (int k = 0; k < 4; ++k) {
            poemv_core(pv[k], xv[k], dv[k], fv[k],
                       omega, rr, lam, m10, m11v, m20, m21,
                       a00, a01, a11, b00, b01, b11,
                       ph00, ph01, ph10, ph11, c11, c12, c22,
                       &mean8[2 * k], &cov16[4 * k]);
        }

        v4f m0 = {mean8[0], mean8[1], mean8[2], mean8[3]};
        v4f m1 = {mean8[4], mean8[5], mean8[6], mean8[7]};
        v4f c0 = {cov16[0],  cov16[1],  cov16[2],  cov16[3]};
        v4f c1 = {cov16[4],  cov16[5],  cov16[6],  cov16[7]};
        v4f c2 = {cov16[8],  cov16[9],  cov16[10], cov16[11]};
        v4f c3 = {cov16[12], cov16[13], cov16[14], cov16[15]};

        v4f* mo = (v4f*)(mean_out + (size_t)v * 8);
        __builtin_nontemporal_store(m0, mo);
        __builtin_nontemporal_store(m1, mo + 1);
        v4f* co = (v4f*)(cov_out + (size_t)v * 16);
        __builtin_nontemporal_store(c0, co);
        __builtin_nontemporal_store(c1, co + 1);
        __builtin_nontemporal_store(c2, co + 2);
        __builtin_nontemporal_store(c3, co + 3);

        v = vn;
        valid = nvalid;
        buf ^= 1;
    }
}

// Scalar tail for B % 4 (not hit for B = 4,000,000 but kept for generality).
__global__ void poemv_tail_kernel(
    const float* __restrict__ x,  const float* __restrict__ p,
    const float* __restrict__ dl, const float* __restrict__ f,
    const float* __restrict__ omega_p,
    const float* __restrict__ mu1, const float* __restrict__ mu2,
    const float* __restrict__ S1,  const float* __restrict__ S2,
    const float* __restrict__ phi, const float* __restrict__ ctu,
    const int* __restrict__ lam_p, const int* __restrict__ r_p,
    float* __restrict__ mean_out, float* __restrict__ cov_out,
    int start, int Bn)
{
    int i = start + blockIdx.x * blockDim.x + threadIdx.x;
    if (i >= Bn) return;
    const float e0  = __expf(ctu[0]);
    const float e1  = __expf(ctu[3]);
    const float t10 = ctu[2];
    float mean2[2], cov4[4];
    poemv_core(p[i], x[i], dl[i], f[i],
               omega_p[0], (float)r_p[0], (float)lam_p[0],
               mu1[0], mu1[1], mu2[0], mu2[1],
               S1[0], S1[1], S1[3], S2[0], S2[1], S2[3],
               phi[0], phi[1], phi[2], phi[3],
               e0 * e0, e0 * t10, t10 * t10 + e1 * e1,
               mean2, cov4);
    mean_out[2 * (size_t)i + 0] = mean2[0];
    mean_out[2 * (size_t)i + 1] = mean2[1];
    cov_out[4 * (size_t)i + 0] = cov4[0];
    cov_out[4 * (size_t)i + 1] = cov4[1];
    cov_out[4 * (size_t)i + 2] = cov4[2];
    cov_out[4 * (size_t)i + 3] = cov4[3];
}

extern "C" void kernel_launch(void* const* d_in, const int* in_sizes, int n_in,
                              void* d_out, int out_size, void* d_ws, size_t ws_size,
                              hipStream_t stream) {
    const float* x     = (const float*)d_in[0];
    const float* omega = (const float*)d_in[1];
    const float* p     = (const float*)d_in[2];
    const float* dl    = (const float*)d_in[3];
    const float* f     = (const float*)d_in[4];
    const float* mu1   = (const float*)d_in[5];
    const float* mu2   = (const float*)d_in[6];
    const float* S1    = (const float*)d_in[7];
    const float* S2    = (const float*)d_in[8];
    const float* phi   = (const float*)d_in[9];
    const float* ctu   = (const float*)d_in[10];
    const int*   lam   = (const int*)d_in[11];
    const int*   r     = (const int*)d_in[12];

    const int Bn   = in_sizes[0];
    const int nvec = Bn / 4;
    const int rem  = Bn - nvec * 4;

    float* mean_out = (float*)d_out;
    float* cov_out  = mean_out + 2ull * (size_t)Bn;

    if (nvec > 0) {
        int blocks = (nvec + 255) / 256;
        if (blocks > 2048) blocks = 2048;
        poemv_vec_kernel<<<blocks, 256, 0, stream>>>(
            x, p, dl, f, omega, mu1, mu2, S1, S2, phi, ctu, lam, r,
            mean_out, cov_out, nvec);
    }
    if (rem > 0) {
        poemv_tail_kernel<<<1, 64, 0, stream>>>(
            x, p, dl, f, omega, mu1, mu2, S1, S2, phi, ctu, lam, r,
            mean_out, cov_out, nvec * 4, Bn);
    }
}